// GINe_8830452760941
// MI455X (gfx1250) — compile-verified
//
#include <hip/hip_runtime.h>
#include <math.h>

typedef float v2f __attribute__((ext_vector_type(2)));
typedef float v8f __attribute__((ext_vector_type(8)));

// D = A(16x4 f32) * B(4x16 f32) + C(16x16 f32): exact fp32 WMMA on CDNA5.
#define WMMA_F32(a, b, c) \
  __builtin_amdgcn_wmma_f32_16x16x4_f32(false, (a), false, (b), (short)0, (c), false, false)

static __device__ __forceinline__ v2f ld2(const char* base, unsigned byteoff) {
  return *(const v2f*)(base + byteoff);
}

// ---------------------------------------------------------------------------
// Repack W[K,N] (row-major) -> Wp[pk/4][pn][4] with zero fill.
// Lane (half,l15) then reads its B pair {W[k0+2h][n], W[k0+2h+1][n]} as one
// contiguous 8-byte load at Wp[(k0/4)*pn*4 + n*4 + 2h].
// ---------------------------------------------------------------------------
__global__ void pack_w_k(const float* __restrict__ W, float* __restrict__ Wp,
                         int K, int N, int pk, int pn) {
  const int i = blockIdx.x * blockDim.x + threadIdx.x;
  const int total = pk * pn;
  if (i >= total) return;
  const int j = i & 3;
  const int rest = i >> 2;
  const int n = rest % pn;
  const int g = rest / pn;
  const int k = g * 4 + j;
  Wp[i] = (k < K && n < N) ? W[k * N + n] : 0.f;
}

// ---------------------------------------------------------------------------
// Dense GEMM, one wave -> 16x32 output (two 16x16 WMMA tiles, shared A loads).
// K must be a multiple of 4; A must be valid (real or zero-padded) for k<K.
// MODE 0: C = act(acc+bias); zero written to pad cols N<=ncol<nstore.
// MODE 1: C += scale*(acc+bias).
// Branch-free MAC loop: clamped row, padded weights, byte offsets so the
// backend can use SADDR+u32-voffset global addressing.
// ---------------------------------------------------------------------------
template <int MODE, int RELU>
__global__ __launch_bounds__(32) void wmma_gemm_k(
    const float* __restrict__ A, int lda,
    const float* __restrict__ Wp, int pn,
    const float* __restrict__ bias,
    float* __restrict__ C, int ldc,
    int M, int K, int N, int nstore, float scale) {
  const int lane = threadIdx.x;
  const int half = lane >> 4;
  const int l15  = lane & 15;
  const int m0 = blockIdx.x * 16;
  const int n0 = blockIdx.y * 32;

  int row = m0 + l15;
  row = row < M ? row : (M - 1);  // clamp: loads stay valid, masked at store
  const char* Ab = (const char*)A;
  const char* Wb = (const char*)Wp;
  unsigned aoff = ((unsigned)row * (unsigned)lda + 2u * (unsigned)half) * 4u;
  unsigned boff = ((unsigned)(n0 + l15) * 4u + 2u * (unsigned)half) * 4u;
  const unsigned bstep = (unsigned)pn * 16u;

  v8f acc0 = {};
  v8f acc1 = {};
  const int steps = K >> 2;
#pragma unroll 4
  for (int s = 0; s < steps; ++s) {
    const v2f a  = ld2(Ab, aoff);
    const v2f b0 = ld2(Wb, boff);
    const v2f b1 = ld2(Wb, boff + 256u);
    aoff += 16u;
    boff += bstep;
    acc0 = WMMA_F32(a, b0, acc0);
    acc1 = WMMA_F32(a, b1, acc1);
  }

#pragma unroll
  for (int t = 0; t < 2; ++t) {
    const int ncol = n0 + 16 * t + l15;
    if (ncol >= nstore) continue;
    const bool colOK = ncol < N;
    const float bv = colOK ? bias[ncol] : 0.f;
    const v8f& acc = t ? acc1 : acc0;
#pragma unroll
    for (int r = 0; r < 8; ++r) {
      const int mr = m0 + r + 8 * half;  // lanes 16-31 hold rows r+8
      if (mr < M) {
        float v = 0.f;
        if (colOK) {
          v = acc[r] + bv;
          if (RELU) v = fmaxf(v, 0.f);
        }
        const unsigned off = (unsigned)mr * (unsigned)ldc + (unsigned)ncol;
        if (MODE == 0) C[off] = v;
        else C[off] = C[off] + scale * v;
      }
    }
  }
}

// ---------------------------------------------------------------------------
// Fused-concat GEMM over edges: A_row(edge) = [h[src], h[dst], e[edge]], K=3H.
// Three branch-free sub-loops (one per segment); B pointer runs continuously.
// RELU_H applies relu to the two h segments only.
// ---------------------------------------------------------------------------
template <int RELU_H, int RELU_OUT>
__global__ __launch_bounds__(32) void wmma_gemm_cat_k(
    const float* __restrict__ h, const float* __restrict__ e,
    const int* __restrict__ src, const int* __restrict__ dst,
    const float* __restrict__ Wp, int pn,
    const float* __restrict__ bias,
    float* __restrict__ C, int ldc,
    int M, int Hd, int N, int nstore) {
  const int lane = threadIdx.x;
  const int half = lane >> 4;
  const int l15  = lane & 15;
  const int m0 = blockIdx.x * 16;
  const int n0 = blockIdx.y * 32;

  int row = m0 + l15;
  row = row < M ? row : (M - 1);
  const unsigned lanek = 8u * (unsigned)half;  // byte offset of this half's K pair
  const unsigned hs  = (unsigned)src[row] * (unsigned)Hd * 4u + lanek;
  const unsigned hdd = (unsigned)dst[row] * (unsigned)Hd * 4u + lanek;
  const unsigned eb  = (unsigned)row * (unsigned)Hd * 4u + lanek;

  const char* hb = (const char*)h;
  const char* ebp = (const char*)e;
  const char* Wb = (const char*)Wp;
  unsigned boff = ((unsigned)(n0 + l15) * 4u + 2u * (unsigned)half) * 4u;
  const unsigned bstep = (unsigned)pn * 16u;
  const int steps = Hd >> 2;  // Hd % 4 == 0

  v8f acc0 = {};
  v8f acc1 = {};

  {  // segment 1: h[src]
    unsigned ao = hs;
#pragma unroll 4
    for (int s = 0; s < steps; ++s) {
      v2f a = ld2(hb, ao);
      if (RELU_H) { a.x = fmaxf(a.x, 0.f); a.y = fmaxf(a.y, 0.f); }
      const v2f b0 = ld2(Wb, boff);
      const v2f b1 = ld2(Wb, boff + 256u);
      ao += 16u;
      boff += bstep;
      acc0 = WMMA_F32(a, b0, acc0);
      acc1 = WMMA_F32(a, b1, acc1);
    }
  }
  {  // segment 2: h[dst]
    unsigned ao = hdd;
#pragma unroll 4
    for (int s = 0; s < steps; ++s) {
      v2f a = ld2(hb, ao);
      if (RELU_H) { a.x = fmaxf(a.x, 0.f); a.y = fmaxf(a.y, 0.f); }
      const v2f b0 = ld2(Wb, boff);
      const v2f b1 = ld2(Wb, boff + 256u);
      ao += 16u;
      boff += bstep;
      acc0 = WMMA_F32(a, b0, acc0);
      acc1 = WMMA_F32(a, b1, acc1);
    }
  }
  {  // segment 3: e[edge]
    unsigned ao = eb;
#pragma unroll 4
    for (int s = 0; s < steps; ++s) {
      const v2f a  = ld2(ebp, ao);
      const v2f b0 = ld2(Wb, boff);
      const v2f b1 = ld2(Wb, boff + 256u);
      ao += 16u;
      boff += bstep;
      acc0 = WMMA_F32(a, b0, acc0);
      acc1 = WMMA_F32(a, b1, acc1);
    }
  }

#pragma unroll
  for (int t = 0; t < 2; ++t) {
    const int ncol = n0 + 16 * t + l15;
    if (ncol >= nstore) continue;
    const bool colOK = ncol < N;
    const float bv = colOK ? bias[ncol] : 0.f;
    const v8f& acc = t ? acc1 : acc0;
#pragma unroll
    for (int r = 0; r < 8; ++r) {
      const int mr = m0 + r + 8 * half;
      if (mr < M) {
        float v = 0.f;
        if (colOK) {
          v = acc[r] + bv;
          if (RELU_OUT) v = fmaxf(v, 0.f);
        }
        C[(unsigned)mr * (unsigned)ldc + (unsigned)ncol] = v;
      }
    }
  }
}

// ---------------------------------------------------------------------------
// Elementwise / scatter / batchnorm helpers
// ---------------------------------------------------------------------------
__global__ void copy_f32_k(float* __restrict__ d, const float* __restrict__ s, long n) {
  const long i = (long)blockIdx.x * blockDim.x + threadIdx.x;
  if (i < n) d[i] = s[i];
}

__global__ void zero_f64_k(double* __restrict__ p, int n) {
  const int i = blockIdx.x * blockDim.x + threadIdx.x;
  if (i < n) p[i] = 0.0;
}

// zin[dst] += relu(h[src] + e)  -- one block per edge, one lane per feature.
__global__ void scatter_msg_k(const float* __restrict__ h, const float* __restrict__ e,
                              const int* __restrict__ src, const int* __restrict__ dst,
                              float* __restrict__ zin, int Hd) {
  const long edge = blockIdx.x;
  const int  f    = threadIdx.x;
  if (f >= Hd) return;
  const long s = src[edge];
  const long d = dst[edge];
  const float m = fmaxf(h[s * Hd + f] + e[edge * (long)Hd + f], 0.f);
  atomicAdd(&zin[d * Hd + f], m);
}

// Per-feature sum / sum-of-squares in fp64 (block partials + fp64 atomics).
__global__ void bn_stats_k(const float* __restrict__ z, double* __restrict__ st,
                           long nN, int Hd) {
  __shared__ double s1[128];
  __shared__ double s2[128];
  const int f = blockIdx.x;
  const int t = threadIdx.x;
  double a = 0.0, b = 0.0;
  for (long r = (long)blockIdx.y * blockDim.x + t; r < nN; r += (long)gridDim.y * blockDim.x) {
    const double v = (double)z[r * Hd + f];
    a += v;
    b += v * v;
  }
  s1[t] = a;
  s2[t] = b;
  __syncthreads();
  for (int o = 64; o > 0; o >>= 1) {
    if (t < o) { s1[t] += s1[t + o]; s2[t] += s2[t + o]; }
    __syncthreads();
  }
  if (t == 0) {
    atomicAdd(&st[f], s1[0]);
    atomicAdd(&st[Hd + f], s2[0]);
  }
}

// h = (h + relu((z - mu) * rsqrt(var+eps) * gamma + beta)) * 0.5
__global__ void bn_apply_k(float* __restrict__ h, const float* __restrict__ z,
                           const double* __restrict__ st,
                           const float* __restrict__ gamma, const float* __restrict__ beta,
                           long nN, int Hd) {
  const long i = (long)blockIdx.x * blockDim.x + threadIdx.x;
  const long total = nN * (long)Hd;
  if (i >= total) return;
  const int f = (int)(i % Hd);
  const double mu = st[f] / (double)nN;
  double var = st[Hd + f] / (double)nN - mu * mu;
  if (var < 0.0) var = 0.0;
  const float inv = 1.0f / sqrtf((float)var + 1e-5f);
  const float zb  = (float)((double)z[i] - mu) * inv * gamma[f] + beta[f];
  h[i] = (h[i] + fmaxf(zb, 0.f)) * 0.5f;
}

// ---------------------------------------------------------------------------
// Launcher
// ---------------------------------------------------------------------------
extern "C" void kernel_launch(void* const* d_in, const int* in_sizes, int n_in,
                              void* d_out, int out_size, void* d_ws, size_t ws_size,
                              hipStream_t stream) {
  const float* x        = (const float*)d_in[0];
  const int*   eidx     = (const int*)d_in[1];
  const float* eattr    = (const float*)d_in[2];
  const float* node_w   = (const float*)d_in[3];
  const float* node_b   = (const float*)d_in[4];
  const float* edge_w   = (const float*)d_in[5];
  const float* edge_b   = (const float*)d_in[6];
  const float* conv_w1  = (const float*)d_in[7];
  const float* conv_b1  = (const float*)d_in[8];
  const float* conv_w2  = (const float*)d_in[9];
  const float* conv_b2  = (const float*)d_in[10];
  const float* bn_gamma = (const float*)d_in[11];
  const float* bn_beta  = (const float*)d_in[12];
  const float* emlp_w1  = (const float*)d_in[13];
  const float* emlp_b1  = (const float*)d_in[14];
  const float* emlp_w2  = (const float*)d_in[15];
  const float* emlp_b2  = (const float*)d_in[16];
  const float* mlp_w1   = (const float*)d_in[17];
  const float* mlp_b1   = (const float*)d_in[18];
  const float* mlp_w2   = (const float*)d_in[19];
  const float* mlp_b2   = (const float*)d_in[20];
  const float* mlp_w3   = (const float*)d_in[21];
  const float* mlp_b3   = (const float*)d_in[22];

  const int  NF = 16, ED = 8;
  const int  nN = in_sizes[0] / NF;   // 50000
  const int  nE = in_sizes[2] / ED;   // 800000
  const int  Hd = in_sizes[4];        // 100
  const int  Ll = in_sizes[8] / Hd;   // 2
  const int  F1 = in_sizes[18];       // 50
  const int  F2 = in_sizes[20];       // 25
  const int  F3 = in_sizes[22];       // 2
  const int* srcI = eidx;
  const int* dstI = eidx + nE;

  auto p4  = [](int v) { return (v + 3) & ~3; };
  auto p32 = [](int v) { return (v + 31) & ~31; };
  const int pnH  = p32(Hd);   // 128
  const int pnF1 = p32(F1);   // 64
  const int pnF2 = p32(F2);   // 32
  const int pnF3 = p32(F3);   // 32
  const int kF1  = p4(F1);    // 52 (padded ldc of o1)
  const int kF2  = p4(F2);    // 28 (padded ldc of o2)

  // Workspace carve (256B aligned).
  char* p = (char*)d_ws;
  auto carve = [&](size_t bytes) -> char* {
    char* q = p;
    p += (bytes + 255) & ~(size_t)255;
    return q;
  };
  float*  h   = (float*)carve((size_t)nN * Hd * sizeof(float));
  float*  e   = (float*)carve((size_t)nE * Hd * sizeof(float));
  float*  zin = (float*)carve((size_t)nN * Hd * sizeof(float));
  float*  t   = (float*)carve((size_t)nN * Hd * sizeof(float));
  float*  z   = (float*)carve((size_t)nN * Hd * sizeof(float));
  float*  hid = (float*)carve((size_t)nE * Hd * sizeof(float));
  double* st  = (double*)carve((size_t)2 * Hd * sizeof(double));

  // Repack all weights into padded WMMA-native layout (tiny, done every call).
  auto pack = [&](const float* W, int K, int N) -> float* {
    const int pk = (K + 3) & ~3;
    const int pn = (N + 31) & ~31;
    float* Wp = (float*)carve((size_t)pk * pn * sizeof(float));
    const int total = pk * pn;
    pack_w_k<<<(total + 255) / 256, 256, 0, stream>>>(W, Wp, K, N, pk, pn);
    return Wp;
  };
  float* wp_node = pack(node_w, NF, Hd);
  float* wp_edge = pack(edge_w, ED, Hd);
  float* wp_c1[8];
  float* wp_c2[8];
  float* wp_e1[8];
  float* wp_e2[8];
  for (int i = 0; i < Ll && i < 8; ++i) {
    wp_c1[i] = pack(conv_w1 + (size_t)i * Hd * Hd, Hd, Hd);
    wp_c2[i] = pack(conv_w2 + (size_t)i * Hd * Hd, Hd, Hd);
    wp_e1[i] = pack(emlp_w1 + (size_t)i * 3 * Hd * Hd, 3 * Hd, Hd);
    wp_e2[i] = pack(emlp_w2 + (size_t)i * Hd * Hd, Hd, Hd);
  }
  float* wp_m1 = pack(mlp_w1, 3 * Hd, F1);
  float* wp_m2 = pack(mlp_w2, F1, F2);
  float* wp_m3 = pack(mlp_w3, F2, F3);

  const dim3 blk(32);
  auto grid = [](int M, int nstore) {
    return dim3((unsigned)((M + 15) / 16), (unsigned)((nstore + 31) / 32));
  };

  // Embeddings: h = x@node_w + b ; e = edge_attr@edge_w + b
  wmma_gemm_k<0, 0><<<grid(nN, Hd), blk, 0, stream>>>(
      x, NF, wp_node, pnH, node_b, h, Hd, nN, p4(NF), Hd, Hd, 1.f);
  wmma_gemm_k<0, 0><<<grid(nE, Hd), blk, 0, stream>>>(
      eattr, ED, wp_edge, pnH, edge_b, e, Hd, nE, p4(ED), Hd, Hd, 1.f);

  const long nh = (long)nN * Hd;
  for (int i = 0; i < Ll && i < 8; ++i) {
    // zin = h + segment_sum(relu(h[src]+e), dst)
    copy_f32_k<<<(unsigned)((nh + 255) / 256), 256, 0, stream>>>(zin, h, nh);
    scatter_msg_k<<<(unsigned)nE, 128, 0, stream>>>(h, e, srcI, dstI, zin, Hd);
    // conv MLP
    wmma_gemm_k<0, 1><<<grid(nN, Hd), blk, 0, stream>>>(
        zin, Hd, wp_c1[i], pnH, conv_b1 + i * Hd, t, Hd, nN, Hd, Hd, Hd, 1.f);
    wmma_gemm_k<0, 0><<<grid(nN, Hd), blk, 0, stream>>>(
        t, Hd, wp_c2[i], pnH, conv_b2 + i * Hd, z, Hd, nN, Hd, Hd, Hd, 1.f);
    // batchnorm (train-mode batch stats) + residual
    zero_f64_k<<<1, 256, 0, stream>>>(st, 2 * Hd);
    bn_stats_k<<<dim3(Hd, 16), 128, 0, stream>>>(z, st, nN, Hd);
    bn_apply_k<<<(unsigned)((nh + 255) / 256), 256, 0, stream>>>(
        h, z, st, bn_gamma + i * Hd, bn_beta + i * Hd, nN, Hd);
    // edge update: e += 0.5*(relu([h_s,h_d,e]@W1+b1)@W2 + b2)
    wmma_gemm_cat_k<0, 1><<<grid(nE, Hd), blk, 0, stream>>>(
        h, e, srcI, dstI, wp_e1[i], pnH, emlp_b1 + i * Hd, hid, Hd, nE, Hd, Hd, Hd);
    wmma_gemm_k<1, 0><<<grid(nE, Hd), blk, 0, stream>>>(
        hid, Hd, wp_e2[i], pnH, emlp_b2 + i * Hd, e, Hd, nE, Hd, Hd, Hd, 0.5f);
  }

  // Final classifier over edges; o1/o2 reuse hid (kF1 + kF2 <= Hd columns).
  float* o1 = hid;                       // [E, kF1] (cols F1..kF1-1 zeroed)
  float* o2 = hid + (size_t)nE * kF1;    // [E, kF2] (cols F2..kF2-1 zeroed)
  wmma_gemm_cat_k<1, 1><<<grid(nE, kF1), blk, 0, stream>>>(
      h, e, srcI, dstI, wp_m1, pnF1, mlp_b1, o1, kF1, nE, Hd, F1, kF1);
  wmma_gemm_k<0, 1><<<grid(nE, kF2), blk, 0, stream>>>(
      o1, kF1, wp_m2, pnF2, mlp_b2, o2, kF2, nE, kF1, F2, kF2, 1.f);
  wmma_gemm_k<0, 0><<<grid(nE, F3), blk, 0, stream>>>(
      o2, kF2, wp_m3, pnF3, mlp_b3, (float*)d_out, F3, nE, kF2, F3, F3, 1.f);
}